// LagAttentionTVARFast_44452911514478
// MI455X (gfx1250) — compile-verified
//
#include <hip/hip_runtime.h>
#include <cstdint>

// ---------------------------------------------------------------------------
// LagAttentionTVAR fused kernel for MI455X (gfx1250, wave32).
//
// Math folding (exact, linear algebra only):
//   CW = ctx_w @ conv_w           (D x 9)     cb = ctx_w@conv_b + ctx_b
//   W2 = wq.T @ wk                (D x D)
//   QW = CW.T @ W2                (9 x D)     qb = cb @ W2
//   aW[k] = QW[k]·val_w           ab = qb·val_w          (alpha 9-tap conv)
//   BW[k][l] = QW[k]·base[l]      bb[l] = qb·base[l]     (beta  9-tap conv)
//   hW[k] = CW[:,k]·bias_w        hb = cb·bias_w + bias_b (bias 9-tap conv)
// Then per (b,t,l):
//   logits = x[t-1-l]*alpha[t] + (Xwin · BW[:,l] + bb[l])
// beta via V_WMMA_F32_16X16X4_F32 (M=16 t, N=16 l, K=12 = 9 taps padded,
// 3 chained WMMAs, bb seeded into the accumulator). BW||bb staged into LDS
// by the Tensor Data Mover (TENSOR_LOAD_TO_LDS + s_wait_tensorcnt).
// ---------------------------------------------------------------------------

typedef float    v2f __attribute__((ext_vector_type(2)));
typedef float    v8f __attribute__((ext_vector_type(8)));
typedef unsigned v4u __attribute__((ext_vector_type(4)));
typedef int      v4i_t __attribute__((ext_vector_type(4)));
typedef int      v8i_t __attribute__((ext_vector_type(8)));

#define B_ 8
#define T_ 4096
#define L_ 256
#define D_ 128
#define KC_ 9
#define KPAD 16
#define TOPK_ 8

#define TBLOCK 128      // timesteps per workgroup (8 waves x 16 t)
#define WIN 400         // x window floats in LDS (392 used, padded)
#define LSTR 260        // padded logits row stride (1040 B, 16B-aligned rows)

// workspace float offsets (BW and bb intentionally contiguous for one TDM op)
enum : int {
  OFF_W2 = 0,          // 128*128
  OFF_CW = 16384,      // 128*16 (k-padded, zero for k>=9)
  OFF_CB = 18432,      // 128
  OFF_QW = 18560,      // 16*128 (rows >=9 are zero)
  OFF_QB = 20608,      // 128
  OFF_HW = 20736,      // 16
  OFF_HB = 20752,      // 1
  OFF_BW = 20864,      // 16*256 row-major [k][l]   (byte offset 83456, 16B aligned)
  OFF_BB = 24960,      // 256  (== OFF_BW + 4096, contiguous with BW)
  OFF_AW = 25216,      // 16
  OFF_AB = 25232       // 1
};

#define TDM_BYTES ((KPAD * L_ + L_) * 4)     // 17408 B = BW + bb
#define TDM_ELEMS (TDM_BYTES / 8)            // 2176 x 8-byte elements

// ---- wave32 XOR-swizzle reductions (ds_swizzle_b32: offset = xor<<10 | 0x1f)
__device__ __forceinline__ float wred_max(float v) {
  v = fmaxf(v, __int_as_float(__builtin_amdgcn_ds_swizzle(__float_as_int(v), (16 << 10) | 0x1f)));
  v = fmaxf(v, __int_as_float(__builtin_amdgcn_ds_swizzle(__float_as_int(v), (8  << 10) | 0x1f)));
  v = fmaxf(v, __int_as_float(__builtin_amdgcn_ds_swizzle(__float_as_int(v), (4  << 10) | 0x1f)));
  v = fmaxf(v, __int_as_float(__builtin_amdgcn_ds_swizzle(__float_as_int(v), (2  << 10) | 0x1f)));
  v = fmaxf(v, __int_as_float(__builtin_amdgcn_ds_swizzle(__float_as_int(v), (1  << 10) | 0x1f)));
  return v;
}
__device__ __forceinline__ float wred_sum(float v) {
  v += __int_as_float(__builtin_amdgcn_ds_swizzle(__float_as_int(v), (16 << 10) | 0x1f));
  v += __int_as_float(__builtin_amdgcn_ds_swizzle(__float_as_int(v), (8  << 10) | 0x1f));
  v += __int_as_float(__builtin_amdgcn_ds_swizzle(__float_as_int(v), (4  << 10) | 0x1f));
  v += __int_as_float(__builtin_amdgcn_ds_swizzle(__float_as_int(v), (2  << 10) | 0x1f));
  v += __int_as_float(__builtin_amdgcn_ds_swizzle(__float_as_int(v), (1  << 10) | 0x1f));
  return v;
}

// ---------------- precompute 1: W2, CW, cb -------------------------------
__global__ __launch_bounds__(256) void precompute1(
    const float* __restrict__ wq, const float* __restrict__ wk,
    const float* __restrict__ ctx_w, const float* __restrict__ conv_w,
    const float* __restrict__ conv_b, const float* __restrict__ ctx_b,
    float* __restrict__ ws) {
  int tid = blockIdx.x * 256 + threadIdx.x;
  if (tid < D_ * D_) {
    int d = tid >> 7, j = tid & 127;
    float s = 0.f;
    for (int i = 0; i < D_; ++i) s = fmaf(wq[i * D_ + d], wk[i * D_ + j], s);
    ws[OFF_W2 + d * D_ + j] = s;
  }
  if (blockIdx.x == 0) {
    for (int idx = threadIdx.x; idx < D_ * KPAD; idx += 256) {
      int d = idx >> 4, k = idx & 15;
      float s = 0.f;
      if (k < KC_)
        for (int e = 0; e < D_; ++e) s = fmaf(ctx_w[d * D_ + e], conv_w[e * KC_ + k], s);
      ws[OFF_CW + idx] = s;
    }
    if (threadIdx.x < D_) {
      int d = threadIdx.x;
      float s = ctx_b[d];
      for (int e = 0; e < D_; ++e) s = fmaf(ctx_w[d * D_ + e], conv_b[e], s);
      ws[OFF_CB + d] = s;
    }
  }
}

// ---------------- precompute 2: QW, qb, hW, hb ---------------------------
__global__ __launch_bounds__(256) void precompute2(
    const float* __restrict__ bias_w, const float* __restrict__ bias_b,
    float* __restrict__ ws) {
  int tid = blockIdx.x * 256 + threadIdx.x;
  const float* W2 = ws + OFF_W2;
  const float* CW = ws + OFF_CW;
  const float* cb = ws + OFF_CB;
  if (tid < KPAD * D_) {
    int k = tid >> 7, j = tid & 127;
    float s = 0.f;
    for (int d = 0; d < D_; ++d) s = fmaf(CW[d * 16 + k], W2[d * D_ + j], s);
    ws[OFF_QW + tid] = s;                       // rows k>=9 auto-zero
  } else if (tid < KPAD * D_ + D_) {
    int j = tid - KPAD * D_;
    float s = 0.f;
    for (int d = 0; d < D_; ++d) s = fmaf(cb[d], W2[d * D_ + j], s);
    ws[OFF_QB + j] = s;
  } else if (tid < KPAD * D_ + D_ + KPAD) {
    int k = tid - (KPAD * D_ + D_);
    float s = 0.f;
    for (int d = 0; d < D_; ++d) s = fmaf(CW[d * 16 + k], bias_w[d], s);
    ws[OFF_HW + k] = s;
  } else if (tid == KPAD * D_ + D_ + KPAD) {
    float s = bias_b[0];
    for (int d = 0; d < D_; ++d) s = fmaf(cb[d], bias_w[d], s);
    ws[OFF_HB] = s;
  }
}

// ---------------- precompute 3: BW, bb, aW, ab ---------------------------
__global__ __launch_bounds__(256) void precompute3(
    const float* __restrict__ lag_embed, const float* __restrict__ val_w,
    const float* __restrict__ val_b, float* __restrict__ ws) {
  int tid = blockIdx.x * 256 + threadIdx.x;
  const float* QW = ws + OFF_QW;
  const float* qb = ws + OFF_QB;
  if (tid < KPAD * L_) {
    int k = tid >> 8, l = tid & 255;
    float s = 0.f;
    for (int j = 0; j < D_; ++j)
      s = fmaf(QW[k * D_ + j], val_b[j] + lag_embed[(l + 1) * D_ + j], s);
    ws[OFF_BW + tid] = s;
  } else if (tid < KPAD * L_ + L_) {
    int l = tid - KPAD * L_;
    float s = 0.f;
    for (int j = 0; j < D_; ++j)
      s = fmaf(qb[j], val_b[j] + lag_embed[(l + 1) * D_ + j], s);
    ws[OFF_BB + l] = s;
  } else if (tid < KPAD * L_ + L_ + KPAD) {
    int k = tid - (KPAD * L_ + L_);
    float s = 0.f;
    for (int j = 0; j < D_; ++j) s = fmaf(QW[k * D_ + j], val_w[j], s);
    ws[OFF_AW + k] = s;
  } else if (tid == KPAD * L_ + L_ + KPAD) {
    float s = 0.f;
    for (int j = 0; j < D_; ++j) s = fmaf(qb[j], val_w[j], s);
    ws[OFF_AB] = s;
  }
}

// ---------------- main fused kernel --------------------------------------
__global__ __launch_bounds__(256) void lagattn_main(
    const float* __restrict__ x, const float* __restrict__ ws,
    float* __restrict__ out_mu, float* __restrict__ out_w) {
  extern __shared__ float smem[];
  float* xw      = smem;                        // WIN (1600 B, keeps bwL 16B-aligned)
  float* bwL     = smem + WIN;                  // 16*256
  float* bbL     = bwL + KPAD * L_;             // 256  (contiguous: one TDM tile)
  float* alphaL  = bbL + L_;                    // 128
  float* hL      = alphaL + TBLOCK;             // 128
  float* logitsL = hL + TBLOCK;                 // 8*16*LSTR

  const int tid = threadIdx.x;
  const int wg  = blockIdx.x;
  const int b   = wg >> 5;                      // 32 t-blocks per batch
  const int t0  = (wg & 31) * TBLOCK;
  const float* xb = x + (size_t)b * T_;

  __builtin_prefetch(xb + t0, 0, 1);            // global_prefetch_b8

  // ---- TDM: DMA BW||bb (17408 B) global -> LDS, one descriptor ----------
  const bool issuer = (tid == 0);
  if (issuer) {
    unsigned ldsoff = (unsigned)(uintptr_t)bwL;               // addr[31:0] = LDS byte offset
    unsigned long long ga = (unsigned long long)(uintptr_t)(ws + OFF_BW);
    // D# group0: count=1 | lds_addr | global_addr[56:0] | type=2
    v4u g0 = { 1u, ldsoff, (unsigned)ga,
               ((unsigned)((ga >> 32) & 0x01FFFFFFull)) | (2u << 30) };
    // D# group1: data_size=3 (8B); tensor_dim0=tile_dim0=2176; 1-D tile
    v8i_t g1 = { (int)(3u << 16),                 // data_size
                 (int)((unsigned)TDM_ELEMS << 16),// tensor_dim0[15:0] @ bits63:48
                 0,                               // tensor_dim0[31:16], tensor_dim1 lo
                 (int)((unsigned)TDM_ELEMS << 16),// tile_dim0 @ bits127:112
                 0,                               // tile_dim1=0 (unused), tile_dim2=0
                 (int)TDM_ELEMS,                  // tensor_dim0_stride[31:0]
                 0, 0 };
    v4i_t gz4 = { 0, 0, 0, 0 };
    v8i_t gz8 = { 0, 0, 0, 0, 0, 0, 0, 0 };
    // amdgpu-toolchain (clang-23) 6-arg form:
    // (uint32x4 g0, int32x8 g1, int32x4, int32x4, int32x8, i32 cpol)
    __builtin_amdgcn_tensor_load_to_lds(g0, g1, gz4, gz4, gz8, 0);
  }

  // ---- stage x window into LDS (overlaps the TDM transfer) --------------
  for (int i = tid; i < WIN; i += 256) {
    int gt = t0 - 264 + i;                      // window covers t0-264 .. t0+127
    xw[i] = (gt >= 0 && i < 392) ? xb[gt] : 0.f;
  }
  if (issuer) __builtin_amdgcn_s_wait_tensorcnt(0);
  __syncthreads();

  // ---- alpha / bias-head 9-tap convs (wave-uniform split: waves 0-3 / 4-7)
  if (tid < TBLOCK) {
    float s = ws[OFF_AB];
#pragma unroll
    for (int k = 0; k < KC_; ++k) s = fmaf(ws[OFF_AW + k], xw[tid + 256 + k], s);
    alphaL[tid] = s;
  } else {
    int tl = tid - TBLOCK;
    float s = ws[OFF_HB];
#pragma unroll
    for (int k = 0; k < KC_; ++k) s = fmaf(ws[OFF_HW + k], xw[tl + 256 + k], s);
    hL[tl] = s;
  }
  __syncthreads();

  // ---- per-wave WMMA logits: 16 t x 256 l -------------------------------
  const int wv = tid >> 5, lane = tid & 31;
  const int m = lane & 15, hh = lane >> 4;      // M/N = m, K-half = hh
  const int tbase = wv * 16;
  const int abase = tbase + m + 256;            // xw index of xwin k=0 for row m

  // A (16x12 of the K-padded window): K = 4c + vgpr + 2*hh
  v2f A[3];
#pragma unroll
  for (int c = 0; c < 3; ++c) {
    int k0 = 4 * c + 2 * hh;
    A[c].x = (k0     <= 8) ? xw[abase + k0    ] : 0.f;
    A[c].y = (k0 + 1 <= 8) ? xw[abase + k0 + 1] : 0.f;
  }
  float alph[8];
#pragma unroll
  for (int r = 0; r < 8; ++r) alph[r] = alphaL[tbase + r + 8 * hh];

  float* mylog = logitsL + wv * 16 * LSTR;

#pragma unroll 1
  for (int n = 0; n < 16; ++n) {                // 16 lag tiles of 16
    v2f Bv[3];
#pragma unroll
    for (int c = 0; c < 3; ++c) {
      int kr = 4 * c + 2 * hh;
      Bv[c].x = bwL[ kr      * L_ + n * 16 + m];
      Bv[c].y = bwL[(kr + 1) * L_ + n * 16 + m];
    }
    float bbv = bbL[n * 16 + m];
    float lagv[8];                              // issue lag loads before WMMAs
#pragma unroll
    for (int r = 0; r < 8; ++r)
      lagv[r] = xw[tbase + r + 8 * hh + 263 - (n * 16 + m)];   // x[t-1-l]
    v8f C = {bbv, bbv, bbv, bbv, bbv, bbv, bbv, bbv};
    C = __builtin_amdgcn_wmma_f32_16x16x4_f32(false, A[0], false, Bv[0], (short)0, C, false, false);
    C = __builtin_amdgcn_wmma_f32_16x16x4_f32(false, A[1], false, Bv[1], (short)0, C, false, false);
    C = __builtin_amdgcn_wmma_f32_16x16x4_f32(false, A[2], false, Bv[2], (short)0, C, false, false);
#pragma unroll
    for (int r = 0; r < 8; ++r)                 // D: M = r + 8*hh, N = n*16+m
      mylog[(r + 8 * hh) * LSTR + n * 16 + m] = fmaf(lagv[r], alph[r], C[r]);
  }

  // ---- top-8 threshold + softmax + outputs (per wave, 16 t serial) ------
  // selection uses the bijection l = lane*8 + j  => b128 LDS loads, b128 stores
#pragma unroll 1
  for (int tt = 0; tt < 16; ++tt) {
    const float* row = mylog + tt * LSTR;
    float4 p0 = *(const float4*)(row + lane * 8);       // ds_load_b128 x2
    float4 p1 = *(const float4*)(row + lane * 8 + 4);
    float orig[8] = {p0.x, p0.y, p0.z, p0.w, p1.x, p1.y, p1.z, p1.w};
    float v[8];
#pragma unroll
    for (int j = 0; j < 8; ++j) v[j] = orig[j];

    float M1 = 0.f, kth = 0.f;
#pragma unroll
    for (int it = 0; it < TOPK_; ++it) {
      float lm = v[0];
#pragma unroll
      for (int j = 1; j < 8; ++j) lm = fmaxf(lm, v[j]);
      lm = wred_max(lm);
      if (it == 0) M1 = lm;
      kth = lm;                                  // after 8 iters: 8th largest
      if (it < TOPK_ - 1) {
#pragma unroll
        for (int j = 0; j < 8; ++j) v[j] = (v[j] == lm) ? -__builtin_inff() : v[j];
      }
    }

    float e[8], S = 0.f, MU = 0.f;
    int tl = tbase + tt;
#pragma unroll
    for (int j = 0; j < 8; ++j) {
      float lv = orig[j];
      float eg = (lv >= kth) ? __expf(lv - M1) : 0.f;
      float lag = xw[tl + 263 - (lane * 8 + j)];
      e[j] = eg;
      S += eg;
      MU = fmaf(eg, lag, MU);
    }
    S  = wred_sum(S);
    MU = wred_sum(MU);
    float inv = 1.0f / S;
    int tg = t0 + tl;
    float* wrow = out_w + ((size_t)(b * T_ + tg)) * L_ + lane * 8;
    *(float4*)(wrow)     = make_float4(e[0] * inv, e[1] * inv, e[2] * inv, e[3] * inv);
    *(float4*)(wrow + 4) = make_float4(e[4] * inv, e[5] * inv, e[6] * inv, e[7] * inv);
    if (lane == 0) out_mu[b * T_ + tg] = fmaf(MU, inv, hL[tl]);
  }
}

// ---------------------------------------------------------------------------
extern "C" void kernel_launch(void* const* d_in, const int* in_sizes, int n_in,
                              void* d_out, int out_size, void* d_ws, size_t ws_size,
                              hipStream_t stream) {
  (void)in_sizes; (void)n_in; (void)out_size; (void)ws_size;
  const float* x         = (const float*)d_in[0];
  const float* lag_embed = (const float*)d_in[1];
  const float* val_w     = (const float*)d_in[2];
  const float* val_b     = (const float*)d_in[3];
  const float* conv_w    = (const float*)d_in[4];
  const float* conv_b    = (const float*)d_in[5];
  const float* ctx_w     = (const float*)d_in[6];
  const float* ctx_b     = (const float*)d_in[7];
  const float* wq        = (const float*)d_in[8];
  const float* wk        = (const float*)d_in[9];
  const float* bias_w    = (const float*)d_in[10];
  const float* bias_b    = (const float*)d_in[11];

  float* ws     = (float*)d_ws;
  float* out_mu = (float*)d_out;
  float* out_w  = out_mu + B_ * T_;

  precompute1<<<64, 256, 0, stream>>>(wq, wk, ctx_w, conv_w, conv_b, ctx_b, ws);
  precompute2<<<9, 256, 0, stream>>>(bias_w, bias_b, ws);
  precompute3<<<18, 256, 0, stream>>>(lag_embed, val_w, val_b, ws);

  size_t smem = (size_t)(WIN + KPAD * L_ + L_ + TBLOCK + TBLOCK + 8 * 16 * LSTR) * sizeof(float);
  lagattn_main<<<B_ * (T_ / TBLOCK), 256, smem, stream>>>(x, ws, out_mu, out_w);
}